// Aggregator_13546326851764
// MI455X (gfx1250) — compile-verified
//
#include <hip/hip_runtime.h>

typedef __attribute__((ext_vector_type(2))) float v2f;
typedef __attribute__((ext_vector_type(8))) float v8f;

#define N_USERS    50000
#define N_ENTITIES 100000
#define EMB        64
#define N_EDGES    2000000
#define NNZ        2000000
#define SLOPE      0.2f

// workspace layout (float offsets)
#define WS_RAW   0LL                     // 50000*64   = 3,200,000
#define WS_CNT   3200000LL               // 100,000
#define WS_SCORE 3300000LL               // 50000*8    = 400,000
#define WS_LAT   3700000LL               // 8*64       = 512
#define WS_P     3700512LL               // 64*16 P + 16 q = 1040

__global__ void zero_f32(float* __restrict__ p, long long n) {
    long long i = (long long)blockIdx.x * blockDim.x + threadIdx.x;
    if (i < n) p[i] = 0.0f;
}

// --- KG aggregate: neigh = entity_emb[tail] * weight[etype-1]; atomic scatter by head ---
__global__ void kg_agg(const float* __restrict__ ent, const int* __restrict__ eidx,
                       const int* __restrict__ etype, const float* __restrict__ weight,
                       float* __restrict__ sums, float* __restrict__ cnt) {
    long long t = (long long)blockIdx.x * blockDim.x + threadIdx.x;
    long long e = t >> 4;
    int j = (int)(t & 15);
    if (e >= N_EDGES) return;
    int head = eidx[e];
    int tail = eidx[N_EDGES + e];
    int r = etype[e] - 1;
    float4 a = ((const float4*)(ent + (long long)tail * EMB))[j];
    float4 b = ((const float4*)(weight + (long long)r * EMB))[j];
    float* dst = sums + (long long)head * EMB + j * 4;
    atomicAdd(dst + 0, a.x * b.x);
    atomicAdd(dst + 1, a.y * b.y);
    atomicAdd(dst + 2, a.z * b.z);
    atomicAdd(dst + 3, a.w * b.w);
    if (j == 0) atomicAdd(cnt + head, 1.0f);
}

// --- user raw aggregate: interact_vals * entity_emb[col] scattered by row ---
__global__ void user_raw_agg(const float* __restrict__ ent, const int* __restrict__ rows,
                             const int* __restrict__ cols, const float* __restrict__ vals,
                             float* __restrict__ raw) {
    long long t = (long long)blockIdx.x * blockDim.x + threadIdx.x;
    long long e = t >> 4;
    int j = (int)(t & 15);
    if (e >= NNZ) return;
    int row = rows[e];
    int col = cols[e];
    float v = vals[e];
    float4 a = ((const float4*)(ent + (long long)col * EMB))[j];
    float* dst = raw + (long long)row * EMB + j * 4;
    atomicAdd(dst + 0, v * a.x);
    atomicAdd(dst + 1, v * a.y);
    atomicAdd(dst + 2, v * a.z);
    atomicAdd(dst + 3, v * a.w);
}

__global__ void ent_finalize(float* __restrict__ out, const float* __restrict__ cnt) {
    long long i = (long long)blockIdx.x * blockDim.x + threadIdx.x;
    if (i >= (long long)N_ENTITIES * EMB) return;
    int id = (int)(i >> 6);
    out[i] = out[i] / fmaxf(cnt[id], 1.0f);
}

// --- one-block setup: P[64x16] (+q[16]) and latent_new[8x64] ---
__global__ void setup_small(const float* __restrict__ latent, const float* __restrict__ weight,
                            const float* __restrict__ W1, const float* __restrict__ b1,
                            const float* __restrict__ W2, const float* __restrict__ b2,
                            const float* __restrict__ Wwatt, const float* __restrict__ bwatt,
                            float* __restrict__ Pq, float* __restrict__ lat_ws,
                            float* __restrict__ lat_out) {
    __shared__ float hL[8][64];    // lin(latent, W1, b1)
    __shared__ float hL2[8][64];   // lin(latent, W2, b2)
    __shared__ float hW2[16][64];  // lin(weight, W2, b2)
    int c = threadIdx.x;  // 0..63
    for (int f = 0; f < 8; ++f) {
        float s1 = b1[c], s2 = b2[c];
        for (int k = 0; k < 64; ++k) {
            float le = latent[f * 64 + k];
            s1 += le * W1[c * 64 + k];
            s2 += le * W2[c * 64 + k];
        }
        hL[f][c] = s1;
        hL2[f][c] = s2;
    }
    for (int r = 0; r < 16; ++r) {
        float s = b2[c];
        for (int k = 0; k < 64; ++k) s += weight[r * 64 + k] * W2[c * 64 + k];
        hW2[r][c] = s;
    }
    __syncthreads();
    // P[c][f] = sum_cc W1[cc][c] * hL[f][cc]   (N padded to 16 with zeros)
    for (int f = 0; f < 16; ++f) {
        float s = 0.0f;
        if (f < 8)
            for (int cc = 0; cc < 64; ++cc) s += W1[cc * 64 + c] * hL[f][cc];
        Pq[c * 16 + f] = s;
    }
    // q[f] = sum_cc b1[cc] * hL[f][cc]
    if (c < 16) {
        float s = 0.0f;
        if (c < 8)
            for (int cc = 0; cc < 64; ++cc) s += b1[cc] * hL[c][cc];
        Pq[1024 + c] = s;
    }
    // latent_new: thread f<8 owns row f
    if (c < 8) {
        float sr[16];
        for (int r = 0; r < 16; ++r) {
            float s = 0.0f;
            for (int k = 0; k < 64; ++k) s += hL2[c][k] * hW2[r][k];
            sr[r] = s;
        }
        float y[16], mx = -1e30f;
        for (int r = 0; r < 16; ++r) {
            float s = bwatt[r];
            for (int rr = 0; rr < 16; ++rr) s += sr[rr] * Wwatt[r * 16 + rr];
            s = (s > 0.0f) ? s : SLOPE * s;
            y[r] = s;
            mx = fmaxf(mx, s);
        }
        float d = 0.0f;
        for (int r = 0; r < 16; ++r) { y[r] = expf(y[r] - mx); d += y[r]; }
        float inv = 1.0f / d;
        for (int cc = 0; cc < 64; ++cc) {
            float s = 0.0f;
            for (int r = 0; r < 16; ++r) s += (y[r] * inv) * weight[r * 64 + cc];
            lat_ws[c * 64 + cc] = s;
            lat_out[c * 64 + cc] = s;
        }
    }
}

// --- score_ = user_emb @ P  via V_WMMA_F32_16X16X4_F32 (M=16 users, N=16 pad, K=64) ---
__global__ void score_gemm_wmma(const float* __restrict__ ue, const float* __restrict__ P,
                                float* __restrict__ score_) {
    int wave = (int)((blockIdx.x * blockDim.x + threadIdx.x) >> 5);
    int lane = threadIdx.x & 31;
    if (wave >= N_USERS / 16) return;     // wave-uniform exit; EXEC stays full otherwise
    int n = lane & 15;                    // = row m of A and col n of B
    int half = lane >> 4;
    const float* Arow = ue + ((long long)wave * 16 + n) * EMB;
    v8f acc = {};
    for (int k = 0; k < EMB; k += 4) {
        int kk = k + 2 * half;
        v2f a, b;
        a.x = Arow[kk];
        a.y = Arow[kk + 1];
        b.x = P[kk * 16 + n];
        b.y = P[(kk + 1) * 16 + n];
        acc = __builtin_amdgcn_wmma_f32_16x16x4_f32(
            /*neg_a=*/false, a, /*neg_b=*/false, b,
            /*c_mod=*/(short)0, acc, /*reuse_a=*/false, /*reuse_b=*/false);
    }
    if (n < 8) {
        for (int g = 0; g < 8; ++g) {
            int row = wave * 16 + g + 8 * half;
            score_[(long long)row * 8 + n] = acc[g];
        }
    }
}

// --- per user: (+q) -> lin(W_user_att) -> leaky -> softmax(8), in-place on score ---
__global__ void user_score_softmax(const float* __restrict__ Pq, const float* __restrict__ Wua,
                                   const float* __restrict__ bua, float* __restrict__ score) {
    int u = blockIdx.x * blockDim.x + threadIdx.x;
    if (u >= N_USERS) return;
    float s_[8];
    for (int f = 0; f < 8; ++f) s_[f] = score[(long long)u * 8 + f] + Pq[1024 + f];
    float y[8], mx = -1e30f;
    for (int r = 0; r < 8; ++r) {
        float s = bua[r];
        for (int f = 0; f < 8; ++f) s += s_[f] * Wua[r * 8 + f];
        s = (s > 0.0f) ? s : SLOPE * s;
        y[r] = s;
        mx = fmaxf(mx, s);
    }
    float d = 0.0f;
    for (int r = 0; r < 8; ++r) { y[r] = expf(y[r] - mx); d += y[r]; }
    float inv = 1.0f / d;
    for (int r = 0; r < 8; ++r) score[(long long)u * 8 + r] = y[r] * inv;
}

// --- user_agg = raw * (1 + score @ latent_new) ---
__global__ void user_finalize(const float* __restrict__ raw, const float* __restrict__ score,
                              const float* __restrict__ lat, float* __restrict__ out) {
    __shared__ float L[8 * 64];
    for (int i = threadIdx.x; i < 8 * 64; i += blockDim.x) L[i] = lat[i];
    __syncthreads();
    long long idx = (long long)blockIdx.x * blockDim.x + threadIdx.x;
    if (idx >= (long long)N_USERS * EMB) return;
    int u = (int)(idx >> 6);
    int c = (int)(idx & 63);
    float f = 0.0f;
    for (int r = 0; r < 8; ++r) f += score[(long long)u * 8 + r] * L[r * 64 + c];
    out[idx] = raw[idx] * (1.0f + f);
}

extern "C" void kernel_launch(void* const* d_in, const int* in_sizes, int n_in,
                              void* d_out, int out_size, void* d_ws, size_t ws_size,
                              hipStream_t stream) {
    (void)in_sizes; (void)n_in; (void)out_size; (void)ws_size;
    const float* entity_emb   = (const float*)d_in[0];
    const float* user_emb     = (const float*)d_in[1];
    const float* latent_emb   = (const float*)d_in[2];
    const int*   edge_index   = (const int*)d_in[3];
    const int*   edge_type    = (const int*)d_in[4];
    const int*   irows        = (const int*)d_in[5];
    const int*   icols        = (const int*)d_in[6];
    const float* ivals        = (const float*)d_in[7];
    const float* weight       = (const float*)d_in[8];
    /* d_in[9] entity_cate_set unused */
    const float* W_user_att   = (const float*)d_in[10];
    const float* b_user_att   = (const float*)d_in[11];
    const float* W_weight_att = (const float*)d_in[12];
    const float* b_weight_att = (const float*)d_in[13];
    const float* W1           = (const float*)d_in[14];
    const float* b1           = (const float*)d_in[15];
    const float* W2           = (const float*)d_in[16];
    const float* b2           = (const float*)d_in[17];

    float* out_ent  = (float*)d_out;
    float* out_user = out_ent + (long long)N_ENTITIES * EMB;
    float* out_lat  = out_user + (long long)N_USERS * EMB;

    float* ws    = (float*)d_ws;
    float* raw   = ws + WS_RAW;
    float* cnt   = ws + WS_CNT;
    float* score = ws + WS_SCORE;
    float* lat   = ws + WS_LAT;
    float* Pq    = ws + WS_P;

    // 1) zero accumulators (raw + cnt contiguous in ws; entity region of d_out)
    {
        long long n = WS_CNT + N_ENTITIES;  // raw + cnt
        zero_f32<<<(unsigned)((n + 255) / 256), 256, 0, stream>>>(ws, n);
        long long m = (long long)N_ENTITIES * EMB;
        zero_f32<<<(unsigned)((m + 255) / 256), 256, 0, stream>>>(out_ent, m);
    }
    // 2) KG scatter-sum (+counts)
    {
        long long t = (long long)N_EDGES * 16;
        kg_agg<<<(unsigned)((t + 255) / 256), 256, 0, stream>>>(
            entity_emb, edge_index, edge_type, weight, out_ent, cnt);
    }
    // 3) user COO scatter-sum
    {
        long long t = (long long)NNZ * 16;
        user_raw_agg<<<(unsigned)((t + 255) / 256), 256, 0, stream>>>(
            entity_emb, irows, icols, ivals, raw);
    }
    // 4) entity mean
    {
        long long m = (long long)N_ENTITIES * EMB;
        ent_finalize<<<(unsigned)((m + 255) / 256), 256, 0, stream>>>(out_ent, cnt);
    }
    // 5) small matrices: P, q, latent_new
    setup_small<<<1, 64, 0, stream>>>(latent_emb, weight, W1, b1, W2, b2,
                                      W_weight_att, b_weight_att, Pq, lat, out_lat);
    // 6) score_ = user_emb @ P  (WMMA f32 16x16x4)
    {
        int tiles = N_USERS / 16;                  // 3125
        int waves_per_block = 256 / 32;            // 8
        int blocks = (tiles + waves_per_block - 1) / waves_per_block;
        score_gemm_wmma<<<blocks, 256, 0, stream>>>(user_emb, Pq, score);
    }
    // 7) per-user attention softmax
    user_score_softmax<<<(N_USERS + 255) / 256, 256, 0, stream>>>(
        Pq, W_user_att, b_user_att, score);
    // 8) final user aggregate
    {
        long long m = (long long)N_USERS * EMB;
        user_finalize<<<(unsigned)((m + 255) / 256), 256, 0, stream>>>(
            raw, score, lat, out_user);
    }
}